// FNORCF1d_89541478187715
// MI455X (gfx1250) — compile-verified
//
#include <hip/hip_runtime.h>
#include <math.h>

// FNO + Chebyshev hybrid, fp32, MI455X (gfx1250, wave32).
// GEMM contractions -> V_WMMA_F32_16X16X4_F32; B-operand tiles staged to LDS
// via GLOBAL_LOAD_ASYNC_TO_LDS_B32 (+ s_wait_asynccnt) and consumed as
// pair-interleaved ds_load_b64 fragments.

#define B_     8
#define L_     8192
#define W_     128
#define MODES_ 32
#define NL_    4
#define SEG_   4
#define CM_    8
#define H1_    128
#define LS_    (L_/SEG_)   // 2048
#define NF2_   64          // 2*MODES (cos/sin interleaved)
#define KSPLIT_ 16

typedef __attribute__((ext_vector_type(2))) float v2f;
typedef __attribute__((ext_vector_type(8))) float v8f;

__device__ __forceinline__ float gelu_exact(float x) {
    return 0.5f * x * (1.0f + erff(x * 0.70710678118654752440f));
}

// ---------------------------------------------------------------------------
// 16x16x4 fp32 WMMA (codegen-confirmed on gfx1250).
// Fragments (wave32):
//   A (16x4): lane = (M%16)+16g ; a = {A[M][k0+2g], A[M][k0+2g+1]}
//   B (4x16): lane = (N%16)+16g ; b = {B[k0+2g][N], B[k0+2g+1][N]}
//   C/D: VGPR r -> lanes 0-15 (M=r), lanes 16-31 (M=8+r), N=lane%16
// ---------------------------------------------------------------------------
__device__ __forceinline__ v8f wmma4(v2f a, v2f b, v8f c) {
#if __has_builtin(__builtin_amdgcn_wmma_f32_16x16x4_f32)
    return __builtin_amdgcn_wmma_f32_16x16x4_f32(false, a, false, b,
                                                 (short)0, c, false, false);
#else
    int lane = (int)(threadIdx.x & 31u);
    int lr = lane & 15, hi = lane >> 4;
    #pragma unroll
    for (int r = 0; r < 8; ++r) {
        int m = 8 * hi + r;
        float sum = c[r];
        #pragma unroll
        for (int kk = 0; kk < 4; ++kk) {
            float av = __shfl((kk & 1) ? a.y : a.x, (m & 15) + 16 * (kk >> 1), 32);
            float bv = __shfl((kk & 1) ? b.y : b.x, lr + 16 * (kk >> 1), 32);
            sum += av * bv;
        }
        c[r] = sum;
    }
    return c;
#endif
}

// ---------------------------------------------------------------------------
// CDNA5 async global->LDS copy (ASYNCcnt) + wait.
// LDS destination byte offset = low 32 bits of the generic shared pointer.
// ---------------------------------------------------------------------------
__device__ __forceinline__ void async_ld_b32(unsigned lds_off, const float* gaddr) {
    asm volatile("global_load_async_to_lds_b32 %0, %1, off"
                 :: "v"(lds_off), "v"(gaddr) : "memory");
}
__device__ __forceinline__ void wait_async() {
    asm volatile("s_wait_asynccnt 0" ::: "memory");
}

// Stage `krows` rows (stride L_) x 64 cols from gbase(+n0) into pair-
// interleaved LDS: element (k,n) -> shB[((k>>1)*64+n)*2 + (k&1)], so a
// B fragment (ka even) is one aligned 8-byte LDS read.
__device__ __forceinline__ void stage_tile(const float* gbase, int n0, int krows,
                                           float* shB, int kofs) {
    int tid = threadIdx.x;
    int k = tid >> 6;                       // 0..3
    int n = tid & 63;
    const float* gp = gbase + (size_t)k * L_ + n0 + n;
    int kk = kofs + k;
    unsigned lofs = (unsigned)(uintptr_t)&shB[((kk >> 1) * 64 + n) * 2 + (kk & 1)];
    #pragma unroll 4
    for (int it = 0; it < krows / 4; ++it) {
        async_ld_b32(lofs, gp);
        gp += 4 * (size_t)L_;
        lofs += 256u * 4u;                  // 4 k-rows -> +256 dwords
    }
}

__device__ __forceinline__ v2f ldsB(const float* shB, int ka, int n) {
    return *(const v2f*)&shB[((ka >> 1) * 64 + n) * 2];
}
__device__ __forceinline__ float ldsAt(const float* shB, int k, int n) {
    return shB[((k >> 1) * 64 + n) * 2 + (k & 1)];
}

// ---------------------------------------------------------------------------
// fc0: h[b,c,l] = x[b,l,0]*w[0,c] + x[b,l,1]*w[1,c] + bias[c]
// ---------------------------------------------------------------------------
__global__ void k_fc0(const float* __restrict__ x, const float* __restrict__ w,
                      const float* __restrict__ bia, float* __restrict__ h) {
    int idx = blockIdx.x * blockDim.x + threadIdx.x;
    if (idx >= B_ * W_ * L_) return;
    int l = idx % L_;
    int c = (idx / L_) % W_;
    int b = idx / (L_ * W_);
    const float* xp = x + ((size_t)b * L_ + l) * 2;
    h[idx] = xp[0] * w[c] + xp[1] * w[W_ + c] + bia[c];
}

// basis[k][l] and basisT[l][k]: k=2m -> cos(2*pi*m*l/L), k=2m+1 -> sin
__global__ void k_basis(float* __restrict__ basis, float* __restrict__ basisT) {
    int idx = blockIdx.x * blockDim.x + threadIdx.x;
    if (idx >= NF2_ * L_) return;
    int l = idx % L_;
    int k = idx / L_;
    int m = k >> 1;
    int ph = (m * l) % L_;                 // exact integer phase
    float ang = 6.28318530717958647692f * (float)ph / (float)L_;
    float v = (k & 1) ? sinf(ang) : cosf(ang);
    basis[idx] = v;
    basisT[(size_t)l * NF2_ + k] = v;
}

__global__ void k_zero(float* __restrict__ p, int n) {
    int idx = blockIdx.x * blockDim.x + threadIdx.x;
    if (idx < n) p[idx] = 0.0f;
}

// ---------------------------------------------------------------------------
// Forward truncated DFT as WMMA GEMM: hf[row][k] = sum_l h[row][l]*basisT[l][k]
// (sign-flip on odd k folds the -j of the forward transform). M=1024 rows,
// N=64, K=8192 split 16 ways; partials combined with global_atomic_add_f32.
// ---------------------------------------------------------------------------
__global__ void k_dftw(const float* __restrict__ h, const float* __restrict__ basisT,
                       float* __restrict__ hf) {
    int ks = blockIdx.x;
    int mblk = blockIdx.y;                 // 8 blocks of 128 rows
    int tid = threadIdx.x;
    int wv = tid >> 5, lane = tid & 31;
    int hi = lane >> 4, lr = lane & 15;
    int m0 = mblk * 128 + wv * 16;
    int kbeg = ks * (L_ / KSPLIT_);
    v8f acc[4] = {};
    const float* hrow = h + (size_t)(m0 + lr) * L_;
    for (int k0 = kbeg; k0 < kbeg + L_ / KSPLIT_; k0 += 4) {
        int ka = k0 + 2 * hi;
        v2f a = *(const v2f*)&hrow[ka];
        #pragma unroll
        for (int j = 0; j < 4; ++j) {
            v2f bb;
            bb.x = basisT[(size_t)ka * NF2_ + j * 16 + lr];
            bb.y = basisT[(size_t)(ka + 1) * NF2_ + j * 16 + lr];
            acc[j] = wmma4(a, bb, acc[j]);
        }
    }
    #pragma unroll
    for (int j = 0; j < 4; ++j) {
        #pragma unroll
        for (int r = 0; r < 8; ++r) {
            int row = m0 + 8 * hi + r;
            int col = j * 16 + lr;
            float v = (col & 1) ? -acc[j][r] : acc[j][r];
            atomicAdd(&hf[(size_t)row * NF2_ + col], v);
        }
    }
}

// ---------------------------------------------------------------------------
// Spectral mode mix + irfft coefficients:
//  om[b,o,m] = sum_c hf[b,c,m]*(wr+j wi)[c,o,m]
//  A2[b,o,2m] = Re*s, A2[b,o,2m+1] = -Im*s, s=(m?2:1)/L  (irfft drops Im(DC))
// ---------------------------------------------------------------------------
__global__ void k_spec(const float* __restrict__ hf, const float* __restrict__ wr,
                       const float* __restrict__ wi, float* __restrict__ A2) {
    int m = blockIdx.x, b = blockIdx.y, o = threadIdx.x;   // block 128
    __shared__ float shr[W_], shi[W_];
    shr[o] = hf[(b * W_ + o) * NF2_ + 2 * m];
    shi[o] = hf[(b * W_ + o) * NF2_ + 2 * m + 1];
    __syncthreads();
    float orr = 0.0f, oii = 0.0f;
    for (int c = 0; c < W_; ++c) {
        float wrr = wr[((size_t)c * W_ + o) * MODES_ + m];
        float wii = wi[((size_t)c * W_ + o) * MODES_ + m];
        float ar = shr[c], ai = shi[c];
        orr += ar * wrr - ai * wii;
        oii += ar * wii + ai * wrr;
    }
    float sc = ((m == 0) ? 1.0f : 2.0f) / (float)L_;
    A2[((size_t)b * W_ + o) * NF2_ + 2 * m]     =  orr * sc;
    A2[((size_t)b * W_ + o) * NF2_ + 2 * m + 1] = -oii * sc;
}

// Per-segment global min/max over (b, c, pos). One block per segment.
__global__ void k_minmax(const float* __restrict__ h, float* __restrict__ mn,
                         float* __restrict__ mx) {
    int s = blockIdx.x, tid = threadIdx.x;
    float lmin = 3.402823466e38f, lmax = -3.402823466e38f;
    const int total = B_ * W_ * LS_;
    for (int f = tid; f < total; f += 256) {
        int bc = f >> 11;
        int j  = f & (LS_ - 1);
        float v = h[(size_t)bc * L_ + s * LS_ + j];
        lmin = fminf(lmin, v);
        lmax = fmaxf(lmax, v);
    }
    __shared__ float smn[256], smx[256];
    smn[tid] = lmin; smx[tid] = lmax;
    __syncthreads();
    for (int off = 128; off > 0; off >>= 1) {
        if (tid < off) {
            smn[tid] = fminf(smn[tid], smn[tid + off]);
            smx[tid] = fmaxf(smx[tid], smx[tid + off]);
        }
        __syncthreads();
    }
    if (tid == 0) { mn[s] = smn[0]; mx[s] = smx[0]; }
}

// wbar[s,m,c] = mean_o cheb_w[layer][s][m][c][o]
__global__ void k_wbar(const float* __restrict__ cw, float* __restrict__ wbar) {
    int idx = blockIdx.x * blockDim.x + threadIdx.x;
    if (idx >= SEG_ * CM_ * W_) return;
    const float* p = cw + (size_t)idx * W_;
    float sum = 0.0f;
    for (int o = 0; o < W_; ++o) sum += p[o];
    wbar[idx] = sum * (1.0f / (float)W_);
}

// Chebyshev coeffs + rec, one block per (b,c,s):
//  C[m] = mean_j xn*T_m(xn);  rec = tanh(sum_m C[m]*wbar[s,m,c])
__global__ void k_cheb(const float* __restrict__ h, const float* __restrict__ mn,
                       const float* __restrict__ mx, const float* __restrict__ wbar,
                       float* __restrict__ rec) {
    int blk = blockIdx.x;                  // (b*W + c)*SEG + s
    int s = blk & (SEG_ - 1);
    int row = blk >> 2;
    int c = row & (W_ - 1);
    int tid = threadIdx.x;
    float vmn = mn[s], vmx = mx[s];
    float inv = 2.0f / (vmx - vmn);
    float sums[CM_];
    #pragma unroll
    for (int t = 0; t < CM_; ++t) sums[t] = 0.0f;
    const float* hp = h + (size_t)row * L_ + s * LS_;
    for (int j = tid; j < LS_; j += 256) {
        float xn = (hp[j] - vmn) * inv - 1.0f;
        float Tpp = 1.0f, Tp = xn;
        sums[0] += xn;
        sums[1] += xn * xn;
        #pragma unroll
        for (int t = 2; t < CM_; ++t) {
            float Tn = 2.0f * xn * Tp - Tpp;
            sums[t] += xn * Tn;
            Tpp = Tp; Tp = Tn;
        }
    }
    __shared__ float red[256];
    __shared__ float Cc[CM_];
    for (int t = 0; t < CM_; ++t) {
        red[tid] = sums[t];
        __syncthreads();
        for (int off = 128; off > 0; off >>= 1) {
            if (tid < off) red[tid] += red[tid + off];
            __syncthreads();
        }
        if (tid == 0) Cc[t] = red[0] * (1.0f / (float)LS_);
        __syncthreads();
    }
    if (tid == 0) {
        float acc = 0.0f;
        #pragma unroll
        for (int t = 0; t < CM_; ++t) acc += Cc[t] * wbar[(s * CM_ + t) * W_ + c];
        rec[blk] = tanhf(acc);
    }
}

// g2[b,s,o] = sum_c rec[b,c,s]*gate_w[layer][W+c][o]  (x_cft gate half,
// collapsed: x_cft is piecewise-constant per segment)
__global__ void k_g2(const float* __restrict__ rec, const float* __restrict__ gw2,
                     float* __restrict__ g2) {
    int s = blockIdx.x, b = blockIdx.y, o = threadIdx.x;   // block 128
    __shared__ float r[W_];
    r[o] = rec[(b * W_ + o) * SEG_ + s];
    __syncthreads();
    float acc = 0.0f;
    for (int c = 0; c < W_; ++c) acc += r[c] * gw2[(size_t)c * W_ + o];
    g2[(b * SEG_ + s) * W_ + o] = acc;
}

// ---------------------------------------------------------------------------
// Fused spectral-inverse + 1x1 conv + bias + GELU -> x_fno[b,o,l].
// Tile M=128 x N=64 per workgroup; B tile (K=192 x 64) async-staged to LDS.
// ---------------------------------------------------------------------------
__global__ void k_fno(const float* __restrict__ h, const float* __restrict__ convw,
                      const float* __restrict__ convb, const float* __restrict__ A2,
                      const float* __restrict__ basis, float* __restrict__ xfno) {
    __shared__ float shB[(W_ + NF2_) * 64];        // 48 KB, pair-interleaved
    int b  = blockIdx.y;
    int n0 = blockIdx.x * 64;
    int tid = threadIdx.x;
    int wv = tid >> 5, lane = tid & 31;
    int hi = lane >> 4, lr = lane & 15;
    int m0 = wv * 16;
    const float* hb = h + (size_t)b * W_ * L_;
    stage_tile(hb, n0, W_, shB, 0);
    stage_tile(basis, n0, NF2_, shB, W_);
    wait_async();
    __syncthreads();
    v8f acc[4] = {};
    for (int k0 = 0; k0 < W_; k0 += 4) {           // K over channels
        int ka = k0 + 2 * hi;
        v2f a;                                     // A[o][c] = conv_w[c][o]
        a.x = convw[(size_t)ka * W_ + m0 + lr];
        a.y = convw[(size_t)(ka + 1) * W_ + m0 + lr];
        #pragma unroll
        for (int j = 0; j < 4; ++j)
            acc[j] = wmma4(a, ldsB(shB, ka, j * 16 + lr), acc[j]);
    }
    const float* A2b = A2 + (size_t)b * W_ * NF2_;
    for (int k0 = 0; k0 < NF2_; k0 += 4) {         // K over basis rows
        int ka = k0 + 2 * hi;
        v2f a = *(const v2f*)&A2b[(size_t)(m0 + lr) * NF2_ + ka];
        #pragma unroll
        for (int j = 0; j < 4; ++j)
            acc[j] = wmma4(a, ldsB(shB, W_ + ka, j * 16 + lr), acc[j]);
    }
    float* ob = xfno + (size_t)b * W_ * L_;
    #pragma unroll
    for (int r = 0; r < 8; ++r) {
        int row = m0 + 8 * hi + r;
        float bias = convb[row];
        #pragma unroll
        for (int j = 0; j < 4; ++j)
            ob[(size_t)row * L_ + n0 + j * 16 + lr] = gelu_exact(acc[j][r] + bias);
    }
}

// ---------------------------------------------------------------------------
// Fused gate GEMM + sigmoid + residual, writes h in place:
//  h[b,o,l] = x_fno + sigmoid(sum_c gw1[c][o]*x_fno[c,l] + g2[b,s,o] + gb[o])
//             * rec[b,o,s]          (x_fno re-read from the staged LDS tile)
// ---------------------------------------------------------------------------
__global__ void k_gate(const float* __restrict__ xfno, const float* __restrict__ gw1,
                       const float* __restrict__ gb, const float* __restrict__ g2,
                       const float* __restrict__ rec, float* __restrict__ h) {
    __shared__ float shB[W_ * 64];                 // 32 KB
    int b  = blockIdx.y;
    int n0 = blockIdx.x * 64;
    int tid = threadIdx.x;
    int wv = tid >> 5, lane = tid & 31;
    int hi = lane >> 4, lr = lane & 15;
    int m0 = wv * 16;
    int s = n0 >> 11;                              // segment (tile inside one)
    const float* xb = xfno + (size_t)b * W_ * L_;
    stage_tile(xb, n0, W_, shB, 0);
    wait_async();
    __syncthreads();
    v8f acc[4] = {};
    for (int k0 = 0; k0 < W_; k0 += 4) {
        int ka = k0 + 2 * hi;
        v2f a;
        a.x = gw1[(size_t)ka * W_ + m0 + lr];
        a.y = gw1[(size_t)(ka + 1) * W_ + m0 + lr];
        #pragma unroll
        for (int j = 0; j < 4; ++j)
            acc[j] = wmma4(a, ldsB(shB, ka, j * 16 + lr), acc[j]);
    }
    float* hb = h + (size_t)b * W_ * L_;
    #pragma unroll
    for (int r = 0; r < 8; ++r) {
        int row = m0 + 8 * hi + r;
        float gz = g2[(b * SEG_ + s) * W_ + row] + gb[row];
        float rv = rec[(b * W_ + row) * SEG_ + s];
        #pragma unroll
        for (int j = 0; j < 4; ++j) {
            int nn = j * 16 + lr;
            float g = 1.0f / (1.0f + expf(-(acc[j][r] + gz)));
            float xf = ldsAt(shB, row, nn);
            hb[(size_t)row * L_ + n0 + nn] = xf + g * rv;
        }
    }
}

// ---------------------------------------------------------------------------
// Head: out[b,l] = sum_h1 gelu(sum_c fc1_w[c,h1]*h[c,l]+fc1_b)*fc2_w + fc2_b
// Workgroup owns full H1=128 for 64 columns; in-register 8-row sums,
// __shfl_xor(16) cross-half (wave32), LDS cross-wave reduce.
// ---------------------------------------------------------------------------
__global__ void k_head(const float* __restrict__ h, const float* __restrict__ fc1w,
                       const float* __restrict__ fc1b, const float* __restrict__ fc2w,
                       const float* __restrict__ fc2b, float* __restrict__ out) {
    __shared__ float shB[W_ * 64];                 // 32 KB
    __shared__ float partial[8 * 64];
    int b  = blockIdx.y;
    int n0 = blockIdx.x * 64;
    int tid = threadIdx.x;
    int wv = tid >> 5, lane = tid & 31;
    int hi = lane >> 4, lr = lane & 15;
    int m0 = wv * 16;
    const float* hb = h + (size_t)b * W_ * L_;
    stage_tile(hb, n0, W_, shB, 0);
    wait_async();
    __syncthreads();
    v8f acc[4] = {};
    for (int k0 = 0; k0 < W_; k0 += 4) {
        int ka = k0 + 2 * hi;
        v2f a;
        a.x = fc1w[(size_t)ka * H1_ + m0 + lr];
        a.y = fc1w[(size_t)(ka + 1) * H1_ + m0 + lr];
        #pragma unroll
        for (int j = 0; j < 4; ++j)
            acc[j] = wmma4(a, ldsB(shB, ka, j * 16 + lr), acc[j]);
    }
    #pragma unroll
    for (int j = 0; j < 4; ++j) {
        float sacc = 0.0f;
        #pragma unroll
        for (int r = 0; r < 8; ++r) {
            int row = m0 + 8 * hi + r;
            sacc += gelu_exact(acc[j][r] + fc1b[row]) * fc2w[row];
        }
        sacc += __shfl_xor(sacc, 16, 32);          // lanes 0-15 hold 16-row sum
        if (lane < 16) partial[wv * 64 + j * 16 + lr] = sacc;
    }
    __syncthreads();
    if (tid < 64) {
        float t = fc2b[0];
        #pragma unroll
        for (int w = 0; w < 8; ++w) t += partial[w * 64 + tid];
        out[(size_t)b * L_ + n0 + tid] = t;
    }
}

// ---------------------------------------------------------------------------
extern "C" void kernel_launch(void* const* d_in, const int* in_sizes, int n_in,
                              void* d_out, int out_size, void* d_ws, size_t ws_size,
                              hipStream_t stream) {
    (void)in_sizes; (void)n_in; (void)out_size; (void)ws_size;
    const float* x     = (const float*)d_in[0];
    const float* fc0w  = (const float*)d_in[1];
    const float* fc0b  = (const float*)d_in[2];
    const float* swr   = (const float*)d_in[3];
    const float* swi   = (const float*)d_in[4];
    const float* convw = (const float*)d_in[5];
    const float* convb = (const float*)d_in[6];
    const float* chebw = (const float*)d_in[7];
    const float* gatew = (const float*)d_in[8];
    const float* gateb = (const float*)d_in[9];
    const float* fc1w  = (const float*)d_in[10];
    const float* fc1b  = (const float*)d_in[11];
    const float* fc2w  = (const float*)d_in[12];
    const float* fc2b  = (const float*)d_in[13];
    float* out = (float*)d_out;

    float* ws = (float*)d_ws;
    size_t off = 0;
    float* h      = ws + off; off += (size_t)B_ * W_ * L_;   // 32 MB
    float* xf     = ws + off; off += (size_t)B_ * W_ * L_;   // 32 MB
    float* basis  = ws + off; off += (size_t)NF2_ * L_;      // 2 MB [k][l]
    float* basisT = ws + off; off += (size_t)NF2_ * L_;      // 2 MB [l][k]
    float* hf     = ws + off; off += (size_t)B_ * W_ * NF2_;
    float* A2     = ws + off; off += (size_t)B_ * W_ * NF2_;
    float* mn     = ws + off; off += SEG_;
    float* mx     = ws + off; off += SEG_;
    float* rec    = ws + off; off += (size_t)B_ * W_ * SEG_;
    float* wbar   = ws + off; off += (size_t)SEG_ * CM_ * W_;
    float* g2     = ws + off; off += (size_t)B_ * SEG_ * W_;

    k_basis<<<(NF2_ * L_ + 255) / 256, 256, 0, stream>>>(basis, basisT);
    k_fc0<<<(B_ * W_ * L_ + 255) / 256, 256, 0, stream>>>(x, fc0w, fc0b, h);

    for (int i = 0; i < NL_; ++i) {
        k_zero<<<(B_ * W_ * NF2_ + 255) / 256, 256, 0, stream>>>(hf, B_ * W_ * NF2_);
        k_dftw<<<dim3(KSPLIT_, (B_ * W_) / 128), 256, 0, stream>>>(h, basisT, hf);
        k_spec<<<dim3(MODES_, B_), W_, 0, stream>>>(
            hf, swr + (size_t)i * W_ * W_ * MODES_,
            swi + (size_t)i * W_ * W_ * MODES_, A2);
        k_minmax<<<SEG_, 256, 0, stream>>>(h, mn, mx);
        k_wbar<<<(SEG_ * CM_ * W_ + 255) / 256, 256, 0, stream>>>(
            chebw + (size_t)i * SEG_ * CM_ * W_ * W_, wbar);
        k_cheb<<<B_ * W_ * SEG_, 256, 0, stream>>>(h, mn, mx, wbar, rec);
        k_g2<<<dim3(SEG_, B_), W_, 0, stream>>>(
            rec, gatew + (size_t)i * 2 * W_ * W_ + (size_t)W_ * W_, g2);
        k_fno<<<dim3(L_ / 64, B_), 256, 0, stream>>>(
            h, convw + (size_t)i * W_ * W_, convb + (size_t)i * W_, A2, basis, xf);
        k_gate<<<dim3(L_ / 64, B_), 256, 0, stream>>>(
            xf, gatew + (size_t)i * 2 * W_ * W_, gateb + (size_t)i * W_, g2, rec, h);
    }

    k_head<<<dim3(L_ / 64, B_), 256, 0, stream>>>(h, fc1w, fc1b, fc2w, fc2b, out);
}